// AttentionPooling_89730456748661
// MI455X (gfx1250) — compile-verified
//
#include <hip/hip_runtime.h>
#include <math.h>
#include <stdint.h>

// Problem constants (match the JAX reference).
#define BQ 32
#define LQ 4096
#define HQ 1024
#define TPB 256          // 8 waves (wave32)
#define TILE_R 16        // rows per tile (WMMA M)
#define LDS_STRIDE 1028  // 1024 + 4 floats pad: row r -> bank (4*r)%64, conflict-free

typedef float v2f __attribute__((ext_vector_type(2)));
typedef float v8f __attribute__((ext_vector_type(8)));

// Async global->LDS copy, 16B per lane, tracked by ASYNCcnt (CDNA5 TDM-lite path).
__device__ __forceinline__ void async_copy_b128(const float* gsrc, unsigned lds_byte_addr) {
    asm volatile("global_load_async_to_lds_b128 %0, %1, off"
                 :: "v"(lds_byte_addr), "v"(gsrc)
                 : "memory");
}
__device__ __forceinline__ void wait_async_le16() {
    asm volatile("s_wait_asynccnt 0x10" ::: "memory");
}
__device__ __forceinline__ void wait_async_le0() {
    asm volatile("s_wait_asynccnt 0x0" ::: "memory");
}

// Single-pass online-softmax attention pooling over one (batch, L-chunk).
// 256 threads; thread t owns output columns [4t, 4t+3]. Double-buffered async
// tile loads overlap HBM fetch of tile t+1 with WMMA/accumulate of tile t.
__global__ __launch_bounds__(TPB) void attn_pool_pass1(
    const float* __restrict__ h, const int* __restrict__ mask,
    const float* __restrict__ q, float* __restrict__ out,
    float* __restrict__ part, float* __restrict__ msbuf, int nchunk)
{
    __shared__ __align__(16) float tileS[2][TILE_R * LDS_STRIDE]; // 2 x 64.25 KB
    __shared__ float qS[HQ];                                      // 4 KB query copy
    __shared__ float spart[8 * TILE_R];                           // per-wave partial scores

    const int t    = threadIdx.x;
    const int wave = t >> 5;
    const int lane = t & 31;
    const int half = lane >> 4;   // 0: K={0,1}, 1: K={2,3} of the 16x4 A-frag
    const int mrow = lane & 15;   // A-matrix row this lane feeds

    const int lc     = LQ / nchunk;
    const int b      = blockIdx.x / nchunk;
    const int chunk  = blockIdx.x % nchunk;
    const int l0     = chunk * lc;
    const int ntiles = lc / TILE_R;

    // Stage query into LDS once.
    #pragma unroll
    for (int i = 0; i < HQ / TPB; ++i) qS[i * TPB + t] = q[i * TPB + t];

    const float* hb = h + (size_t)b * LQ * HQ;
    const int*   mb = mask + (size_t)b * LQ + l0;

    float4 acc = make_float4(0.f, 0.f, 0.f, 0.f);
    float  m_run = -INFINITY;
    float  s_run = 0.f;

    // Prologue: kick off async fill of tile 0 into buffer 0.
    #pragma unroll
    for (int r = 0; r < TILE_R; ++r)
        async_copy_b128(hb + (size_t)(l0 + r) * HQ + 4 * t,
                        (unsigned)(uintptr_t)&tileS[0][r * LDS_STRIDE + 4 * t]);

    for (int tile = 0; tile < ntiles; ++tile) {
        const int buf = tile & 1;

        // Issue async fill of the next tile into the other buffer, then wait
        // only for the current tile's 16 copies (in-order completion).
        if (tile + 1 < ntiles) {
            const int lb1 = l0 + (tile + 1) * TILE_R;
            #pragma unroll
            for (int r = 0; r < TILE_R; ++r)
                async_copy_b128(hb + (size_t)(lb1 + r) * HQ + 4 * t,
                                (unsigned)(uintptr_t)&tileS[buf ^ 1][r * LDS_STRIDE + 4 * t]);
            wait_async_le16();
        } else {
            wait_async_le0();
        }
        __syncthreads();  // current tile visible to all waves

        // ---- Scores via V_WMMA_F32_16X16X4_F32 ----
        // Wave w covers K in [w*128, w*128+128): 32 chained WMMAs.
        // B-frag replicates q across all 16 columns -> every column of D = scores.
        v8f c = {};
        const float* arow = &tileS[buf][mrow * LDS_STRIDE];
        int kk = wave * 128 + 2 * half;
        #pragma unroll
        for (int step = 0; step < 32; ++step) {
            v2f a;  a[0]  = arow[kk]; a[1]  = arow[kk + 1];
            v2f bq; bq[0] = qS[kk];   bq[1] = qS[kk + 1];
            c = __builtin_amdgcn_wmma_f32_16x16x4_f32(
                    false, a, false, bq, (short)0, c, false, false);
            kk += 4;
        }
        // Column N==0 of D: lane 0 holds M=j (VGPR j), lane 16 holds M=8+j.
        if (lane == 0) {
            #pragma unroll
            for (int j = 0; j < 8; ++j) spart[wave * TILE_R + j] = c[j];
        } else if (lane == 16) {
            #pragma unroll
            for (int j = 0; j < 8; ++j) spart[wave * TILE_R + 8 + j] = c[j];
        }
        __syncthreads();

        // ---- Online softmax update (replicated per thread; LDS broadcast reads) ----
        float sc[TILE_R];
        float m_new = m_run;
        #pragma unroll
        for (int i = 0; i < TILE_R; ++i) {
            float s = 0.f;
            #pragma unroll
            for (int w = 0; w < 8; ++w) s += spart[w * TILE_R + i];
            s *= 0.03125f;                       // 1/sqrt(1024)
            if (mb[tile * TILE_R + i] == 0) s = -INFINITY;
            sc[i] = s;
            m_new = fmaxf(m_new, s);
        }
        const float resc = __expf(m_run - m_new);
        s_run *= resc;
        acc.x *= resc; acc.y *= resc; acc.z *= resc; acc.w *= resc;
        #pragma unroll
        for (int i = 0; i < TILE_R; ++i) {
            const float wgt = __expf(sc[i] - m_new);
            s_run += wgt;
            const float4 v = *(const float4*)(&tileS[buf][i * LDS_STRIDE + 4 * t]);
            acc.x += wgt * v.x; acc.y += wgt * v.y;
            acc.z += wgt * v.z; acc.w += wgt * v.w;
        }
        m_run = m_new;
        __syncthreads();  // protects buffer being refilled next iteration + spart
    }

    if (nchunk == 1) {
        const float inv = 1.f / s_run;
        float4 o = make_float4(acc.x * inv, acc.y * inv, acc.z * inv, acc.w * inv);
        *(float4*)(out + (size_t)b * HQ + 4 * t) = o;
    } else {
        *(float4*)(part + ((size_t)b * nchunk + chunk) * HQ + 4 * t) = acc;
        if (t == 0) {
            msbuf[((size_t)b * nchunk + chunk) * 2 + 0] = m_run;
            msbuf[((size_t)b * nchunk + chunk) * 2 + 1] = s_run;
        }
    }
}

// Merge NCHUNK flash-style partials per batch and normalize.
__global__ __launch_bounds__(TPB) void attn_pool_combine(
    const float* __restrict__ part, const float* __restrict__ msbuf,
    float* __restrict__ out, int nchunk)
{
    const int b = blockIdx.x;
    const int t = threadIdx.x;

    float M = -INFINITY;
    for (int i = 0; i < nchunk; ++i)
        M = fmaxf(M, msbuf[((size_t)b * nchunk + i) * 2 + 0]);

    float S = 0.f;
    float4 o = make_float4(0.f, 0.f, 0.f, 0.f);
    for (int i = 0; i < nchunk; ++i) {
        const float mi = msbuf[((size_t)b * nchunk + i) * 2 + 0];
        const float si = msbuf[((size_t)b * nchunk + i) * 2 + 1];
        const float e  = __expf(mi - M);
        S += si * e;
        const float4 p = *(const float4*)(part + ((size_t)b * nchunk + i) * HQ + 4 * t);
        o.x += e * p.x; o.y += e * p.y; o.z += e * p.z; o.w += e * p.w;
    }
    const float inv = 1.f / S;
    o.x *= inv; o.y *= inv; o.z *= inv; o.w *= inv;
    *(float4*)(out + (size_t)b * HQ + 4 * t) = o;
}

extern "C" void kernel_launch(void* const* d_in, const int* in_sizes, int n_in,
                              void* d_out, int out_size, void* d_ws, size_t ws_size,
                              hipStream_t stream) {
    const float* h    = (const float*)d_in[0];  // [B, L, H] f32
    const int*   mask = (const int*)d_in[1];    // [B, L] i32
    const float* q    = (const float*)d_in[2];  // [H] f32
    float*       out  = (float*)d_out;          // [B, H] f32
    (void)in_sizes; (void)n_in; (void)out_size;

    // Pick the largest chunk count whose partials fit in the workspace.
    int nchunk = 1;
    for (int nc = 16; nc >= 2; nc >>= 1) {
        const size_t need = ((size_t)BQ * nc * HQ + (size_t)BQ * nc * 2) * sizeof(float);
        if (d_ws != nullptr && need <= ws_size) { nchunk = nc; break; }
    }
    float* part  = (float*)d_ws;
    float* msbuf = part + (size_t)BQ * nchunk * HQ;

    attn_pool_pass1<<<BQ * nchunk, TPB, 0, stream>>>(h, mask, q, out, part, msbuf, nchunk);
    if (nchunk > 1)
        attn_pool_combine<<<BQ, TPB, 0, stream>>>(part, msbuf, out, nchunk);
}